// HMSGNet_472446403323
// MI455X (gfx1250) — compile-verified
//
#include <hip/hip_runtime.h>
#include <math.h>
#include <cstddef>

// HMSGNet forward for MI455X (gfx1250), wave32, fp32 WMMA (V_WMMA_F32_16X16X4_F32).
// GEMM: block = 8 waves covering one 16-row M tile x 8 consecutive 16-col N tiles.
// A chunk (16x64) staged cooperatively in LDS (stride 68 -> conflict-free b64
// fragment reads); B fragments are unconditional global b64 loads (L2-resident
// weights). Out-of-range rows/cols load clamped garbage; stores are guarded
// (WMMA row m / col n only affect D row m / col n, so garbage never leaks).

#define HH  256
#define RR  4
#define KK  8
#define NHH 4
#define HDD 64
#define KC  64

typedef __attribute__((ext_vector_type(2))) float v2f;
typedef __attribute__((ext_vector_type(4))) float v4f;
typedef __attribute__((ext_vector_type(8))) float v8f;

enum { GEMM_ACCUM = 1, GEMM_RELU = 2 };

// ---------------------------------------------------------------- utilities

__device__ __forceinline__ void atomicMaxFloat(float* addr, float val) {
    if (val >= 0.f) atomicMax((int*)addr, __float_as_int(val));
    else            atomicMin((unsigned int*)addr, __float_as_uint(val));
}

__device__ __forceinline__ float blockSum256(float v, float* red) {
    int t = threadIdx.x;
    red[t] = v; __syncthreads();
    for (int s = 128; s > 0; s >>= 1) {
        if (t < s) red[t] += red[t + s];
        __syncthreads();
    }
    float r = red[0]; __syncthreads();
    return r;
}

__global__ void fill_kernel(float* p, size_t n, float v) {
    size_t i = (size_t)blockIdx.x * blockDim.x + threadIdx.x;
    if (i < n) p[i] = v;
}

// ---------------------------------------------------------------- WMMA GEMM
// C[M,Nout] = op( A[M,*](lda) x W[Nout,*](ldw)^T + bias )   (linear: x @ w.T + b)
__global__ void gemm_wmma_f32(const float* __restrict__ A, int lda,
                              const float* __restrict__ W, int ldw,
                              const float* __restrict__ bias,
                              const float* __restrict__ rowScale, int rsStride,
                              float* __restrict__ C, int ldc,
                              int M, int Nout, int Kdim, int flags)
{
    __shared__ float As[16 * 68];                 // 16 rows, stride 68 (bank-spread)
    const int tid  = threadIdx.x;
    const int wave = tid >> 5;
    const int lane = tid & 31;
    const int tilesN  = (Nout + 15) >> 4;
    const int nGroups = (tilesN + 7) >> 3;
    const int mTile = blockIdx.x / nGroups;
    const int nTile = (blockIdx.x % nGroups) * 8 + wave;
    const int m0 = mTile * 16;
    const int n0 = nTile * 16;
    const bool active = (nTile < tilesN);         // wave-uniform

    const int  half = lane >> 4;                  // K sub-pair select (ISA 16x4 layout)
    const int  l    = lane & 15;
    const int  nB   = n0 + l;
    const bool nOK  = active && (nB < Nout);
    const float* Wrow  = W + (size_t)(nOK ? nB : 0) * ldw + half * 2;
    const float* lfrag = &As[l * 68 + half * 2];

    // cooperative A-chunk load mapping: thread t -> row t/16, 4 cols at (t%16)*4
    const int arow = tid >> 4;
    const int acol = (tid & 15) * 4;
    int mload = m0 + arow; if (mload >= M) mload = M - 1;   // clamped: garbage is masked at store
    const float* Aload = A + (size_t)mload * lda + acol;
    float* Astore = &As[arow * 68 + acol];

    v8f acc = {0.f,0.f,0.f,0.f,0.f,0.f,0.f,0.f};
    for (int kc = 0; kc < Kdim; kc += KC) {
        __syncthreads();
        *(v4f*)Astore = *(const v4f*)(Aload + kc);          // coalesced b128
        if (kc + KC < Kdim) __builtin_prefetch(Aload + kc + KC, 0, 1);
        __syncthreads();
        if (active) {
            #pragma unroll
            for (int k = 0; k < KC; k += 4) {
                v2f a = *(const v2f*)(lfrag + k);           // ds_load_b64
                v2f b = *(const v2f*)(Wrow + kc + k);       // global_load_b64
                acc = __builtin_amdgcn_wmma_f32_16x16x4_f32(false, a, false, b,
                                                            (short)0, acc, false, false);
            }
        }
    }
    if (!active) return;

    const int n = n0 + l;
    #pragma unroll
    for (int v = 0; v < 8; ++v) {
        int m = m0 + v + half * 8;                // ISA 16x16 f32 C/D layout
        if (m < M && n < Nout) {
            float val = acc[v];
            if (bias) val += bias[n];
            float* cp = C + (size_t)m * ldc + n;
            if (flags & GEMM_ACCUM) val += *cp;
            if (flags & GEMM_RELU)  val = fmaxf(val, 0.f);
            if (rowScale) val *= rowScale[(size_t)m * rsStride];
            *cp = val;
        }
    }
}

// ------------------------------------------------------------ encoder fuse
__global__ void encoder_finalize(const float* __restrict__ ht, const float* __restrict__ ha,
                                 const float* tg, const float* tb,
                                 const float* ag, const float* ab,
                                 const int* __restrict__ ntype, const float* __restrict__ pos,
                                 const float* p1w, const float* p1b,
                                 const float* p2w, const float* p2b,
                                 const float* temb, float* __restrict__ h, int N)
{
    int n = blockIdx.x, t = threadIdx.x;
    __shared__ float red[256];
    __shared__ float p1[128];
    int nt = ntype[n];
    const float* src = nt == 0 ? ht : ha;
    const float* gg  = nt == 0 ? tg : ag;
    const float* bb  = nt == 0 ? tb : ab;
    float v  = src[(size_t)n * HH + t];
    float mu = blockSum256(v, red) * (1.f / HH);
    float d  = v - mu;
    float var = blockSum256(d * d, red) * (1.f / HH);
    float hv = fmaxf(d * rsqrtf(var + 1e-5f) * gg[t] + bb[t], 0.f);
    if (t < 128) p1[t] = fmaxf(p1w[t] * pos[n] + p1b[t], 0.f);
    __syncthreads();
    float pe = p2b[t];
    for (int j = 0; j < 128; ++j) pe += p2w[t * 128 + j] * p1[j];
    h[(size_t)n * HH + t] = hv + pe + temb[nt * HH + t];
}

// ------------------------------------------------------------- edge kernels
__global__ void edge_score(const int* __restrict__ src, const int* __restrict__ dst,
                           const int* __restrict__ et, int E, int r,
                           const float* __restrict__ Q, const float* __restrict__ Kp,
                           float* __restrict__ score, float* __restrict__ mb)
{
    int t = blockIdx.x * blockDim.x + threadIdx.x;
    if (t >= E * NHH) return;
    int e = t >> 2, hh = t & 3;
    if (et[e] != r) return;
    int sN = src[e], dN = dst[e];
    const float* q = Q  + (size_t)dN * HH + hh * HDD;
    const float* k = Kp + (size_t)sN * HH + hh * HDD;
    float acc = 0.f;
    for (int i = 0; i < HDD; ++i) acc += q[i] * k[i];
    acc *= 0.125f;                                   // 1/sqrt(64)
    score[(size_t)e * NHH + hh] = acc;
    atomicMaxFloat(&mb[dN * NHH + hh], acc);
}

__global__ void edge_exp(const int* __restrict__ dst, const int* __restrict__ et,
                         int E, int r, float* __restrict__ score,
                         const float* __restrict__ mb, float* __restrict__ zb)
{
    int t = blockIdx.x * blockDim.x + threadIdx.x;
    if (t >= E * NHH) return;
    int e = t >> 2, hh = t & 3;
    if (et[e] != r) return;
    int dN = dst[e];
    float m = mb[dN * NHH + hh];
    if (m < -1e30f) m = 0.f;                         // isfinite fixup
    float ex = expf(score[(size_t)e * NHH + hh] - m);
    score[(size_t)e * NHH + hh] = ex;
    atomicAdd(&zb[dN * NHH + hh], ex);
}

__global__ void edge_agg(const int* __restrict__ src, const int* __restrict__ dst,
                         const int* __restrict__ et, int E, int r,
                         const float* __restrict__ score, const float* __restrict__ zb,
                         const float* __restrict__ V, float* __restrict__ agg)
{
    int t = blockIdx.x * blockDim.x + threadIdx.x;
    if (t >= E * NHH) return;
    int e = t >> 2, hh = t & 3;
    if (et[e] != r) return;
    int sN = src[e], dN = dst[e];
    float al = score[(size_t)e * NHH + hh] / zb[dN * NHH + hh];
    const float* v = V + (size_t)sN * HH + hh * HDD;
    float* ap = agg + ((size_t)dN * RR + r) * HH + hh * HDD;
    for (int i = 0; i < HDD; ++i) atomicAdd(&ap[i], al * v[i]);
}

// -------------------------------------------------------- routing / mixing
__global__ void gate_softmax(float* __restrict__ gate, int N) {
    int n = blockIdx.x * blockDim.x + threadIdx.x;
    if (n >= N) return;
    float* g = gate + (size_t)n * KK;
    float m = g[0];
    for (int k = 1; k < KK; ++k) m = fmaxf(m, g[k]);
    float s = 0.f;
    for (int k = 0; k < KK; ++k) { g[k] = expf(g[k] - m); s += g[k]; }
    float inv = 1.f / s;
    for (int k = 0; k < KK; ++k) g[k] *= inv;
}

__global__ void sew_softmax(const float* __restrict__ sel, float* __restrict__ sew) {
    int k = threadIdx.x;
    if (k >= KK) return;
    float m = sel[k * RR];
    for (int r = 1; r < RR; ++r) m = fmaxf(m, sel[k * RR + r]);
    float e[RR], s = 0.f;
    for (int r = 0; r < RR; ++r) { e[r] = expf(sel[k * RR + r] - m); s += e[r]; }
    for (int r = 0; r < RR; ++r) sew[k * RR + r] = e[r] / s;
}

__global__ void sk_combine(const float* __restrict__ agg, const float* __restrict__ sew,
                           int k, float* __restrict__ out, int N)
{
    size_t t = (size_t)blockIdx.x * blockDim.x + threadIdx.x;
    if (t >= (size_t)N * HH) return;
    int n = (int)(t >> 8), hh = (int)(t & 255);
    float s = 0.f;
    for (int r = 0; r < RR; ++r)
        s += sew[k * RR + r] * agg[((size_t)n * RR + r) * HH + hh];
    out[t] = s;
}

__global__ void resid_ln(float* __restrict__ h, const float* __restrict__ hn,
                         const float* g, const float* b)
{
    int n = blockIdx.x, t = threadIdx.x;
    __shared__ float red[256];
    float v  = h[(size_t)n * HH + t] + hn[(size_t)n * HH + t];
    float mu = blockSum256(v, red) * (1.f / HH);
    float d  = v - mu;
    float var = blockSum256(d * d, red) * (1.f / HH);
    h[(size_t)n * HH + t] = d * rsqrtf(var + 1e-5f) * g[t] + b[t];
}

// ------------------------------------------------------------- pooling/head
__global__ void att_scores(const float* __restrict__ h,
                           const float* tw, const float* tb,
                           const float* aw, const float* ab,
                           const int* __restrict__ ntype, const int* __restrict__ batch,
                           float* __restrict__ st, float* __restrict__ sa,
                           float* __restrict__ mT, float* __restrict__ mA, int N)
{
    int n = blockIdx.x * blockDim.x + threadIdx.x;
    if (n >= N) return;
    const float* hr = h + (size_t)n * HH;
    float s1 = tb[0], s2 = ab[0];
    for (int d = 0; d < HH; ++d) { s1 += hr[d] * tw[d]; s2 += hr[d] * aw[d]; }
    st[n] = s1; sa[n] = s2;
    int b = batch[n];
    if (ntype[n] == 0) atomicMaxFloat(&mT[b], s1);
    else               atomicMaxFloat(&mA[b], s2);
}

__global__ void att_z(float* __restrict__ st, float* __restrict__ sa,
                      const float* __restrict__ mT, const float* __restrict__ mA,
                      const int* __restrict__ ntype, const int* __restrict__ batch,
                      float* __restrict__ zT, float* __restrict__ zA, int N)
{
    int n = blockIdx.x * blockDim.x + threadIdx.x;
    if (n >= N) return;
    int b = batch[n];
    if (ntype[n] == 0) {
        float m = mT[b]; if (m < -1e30f) m = 0.f;
        float e = expf(st[n] - m); st[n] = e; atomicAdd(&zT[b], e);
    } else {
        float m = mA[b]; if (m < -1e30f) m = 0.f;
        float e = expf(sa[n] - m); sa[n] = e; atomicAdd(&zA[b], e);
    }
}

__global__ void att_accum(const float* __restrict__ h,
                          const float* __restrict__ st, const float* __restrict__ sa,
                          const float* __restrict__ zT, const float* __restrict__ zA,
                          const int* __restrict__ ntype, const int* __restrict__ batch,
                          float* __restrict__ gcat, int N)
{
    size_t t = (size_t)blockIdx.x * blockDim.x + threadIdx.x;
    if (t >= (size_t)N * HH) return;
    int n = (int)(t >> 8), d = (int)(t & 255);
    int b = batch[n];
    float hv = h[t];
    if (ntype[n] == 0) {
        float w = st[n] / fmaxf(zT[b], 1e-12f);
        atomicAdd(&gcat[b * 512 + d], hv * w);
    } else {
        float w = sa[n] / fmaxf(zA[b], 1e-12f);
        atomicAdd(&gcat[b * 512 + 256 + d], hv * w);
    }
}

__global__ void ln_relu_rows(const float* __restrict__ in, const float* g,
                             const float* be, float* __restrict__ outp)
{
    int b = blockIdx.x, t = threadIdx.x;
    __shared__ float red[256];
    float v  = in[b * HH + t];
    float mu = blockSum256(v, red) * (1.f / HH);
    float d  = v - mu;
    float var = blockSum256(d * d, red) * (1.f / HH);
    outp[b * HH + t] = fmaxf(d * rsqrtf(var + 1e-5f) * g[t] + be[t], 0.f);
}

__global__ void heads_kernel(const float* __restrict__ gf,
                             const float* d1w, const float* d1b,
                             const float* d2w, const float* d2b,
                             const float* symw, const float* symb,
                             const float* phqw, const float* phqb,
                             float* __restrict__ out)
{
    int b = blockIdx.x, t = threadIdx.x;        // 128 threads
    __shared__ float row[256];
    __shared__ float d1[128];
    row[t]       = gf[b * HH + t];
    row[t + 128] = gf[b * HH + t + 128];
    __syncthreads();
    float a = d1b[t];
    for (int j = 0; j < 256; ++j) a += d1w[t * 256 + j] * row[j];
    d1[t] = fmaxf(a, 0.f);
    __syncthreads();
    if (t == 0) {
        float s = d2b[0];
        for (int j = 0; j < 128; ++j) s += d2w[j] * d1[j];
        out[b * (KK + 2)] = s;
    } else if (t <= KK) {
        int k = t - 1;
        float s = symb[k];
        for (int j = 0; j < 256; ++j) s += symw[k * 256 + j] * row[j];
        out[b * (KK + 2) + 1 + k] = s;
    } else if (t == KK + 1) {
        float s = phqb[0];
        for (int j = 0; j < 256; ++j) s += phqw[j] * row[j];
        out[b * (KK + 2) + 1 + KK] = s;
    }
}

// ------------------------------------------------------------------- launch
extern "C" void kernel_launch(void* const* d_in, const int* in_sizes, int n_in,
                              void* d_out, int out_size, void* d_ws, size_t ws_size,
                              hipStream_t stream)
{
    const float* x     = (const float*)d_in[0];
    const int*   eidx  = (const int*)d_in[1];
    const int*   etype = (const int*)d_in[2];
    const int*   ntype = (const int*)d_in[3];
    const float* pos   = (const float*)d_in[4];
    const int*   batch = (const int*)d_in[5];
    const int N = in_sizes[3];
    const int E = in_sizes[2];
    const int B = out_size / (KK + 2);
    const int* src = eidx;
    const int* dst = eidx + E;
    auto P = [&](int i) { return (const float*)d_in[i]; };
    const int TD = 768;

    // scratch layout (floats), ~86.2M floats = ~345 MB
    float* ws = (float*)d_ws;
    size_t off = 0;
    auto alloc = [&](size_t n) { float* p = ws + off; off += n; return p; };
    float* h     = alloc((size_t)N * HH);
    float* hacc  = alloc((size_t)N * HH);      // encoder ht / cross accumulator
    float* tmp1  = alloc((size_t)N * HH);      // encoder ha / rm1 out / s_k
    float* tmp2  = alloc((size_t)N * HH);      // u_k
    float* Qb    = alloc((size_t)N * HH);
    float* Kb    = alloc((size_t)N * HH);
    float* Vb    = alloc((size_t)N * HH);
    float* agg   = alloc((size_t)N * RR * HH);
    float* score = alloc((size_t)E * NHH);
    float* mb    = alloc((size_t)N * NHH);
    float* zb    = alloc((size_t)N * NHH);
    float* gate  = alloc((size_t)N * KK);
    float* sew   = alloc(KK * RR);
    float* stb   = alloc(N);
    float* sab   = alloc(N);
    float* mT    = alloc(B);
    float* zT    = alloc(B);
    float* mA    = alloc(B);
    float* zA    = alloc(B);
    float* gcat  = alloc((size_t)B * 512);
    float* fbuf  = alloc((size_t)B * HH);
    float* gfb   = alloc((size_t)B * HH);
    (void)ws_size; (void)n_in;

    auto fill = [&](float* p, size_t n, float v) {
        fill_kernel<<<(unsigned)((n + 255) / 256), 256, 0, stream>>>(p, n, v);
    };
    auto gemm = [&](const float* A, int lda, const float* W, int ldw,
                    const float* bias, const float* rs, int rss,
                    float* C, int ldc, int M, int Nout, int Kdim, int flags) {
        int tilesM = (M + 15) / 16;
        int nG = (((Nout + 15) / 16) + 7) / 8;
        gemm_wmma_f32<<<tilesM * nG, 256, 0, stream>>>(
            A, lda, W, ldw, bias, rs, rss, C, ldc, M, Nout, Kdim, flags);
    };
    const float NEGINF = -__builtin_inff();
    const int eThreads = E * NHH, eBlocks = (eThreads + 255) / 256;

    // ---- encoders + positional/type embedding
    gemm(x,      2 * TD, P(7),  TD, P(8),  nullptr, 0, hacc, HH, N, HH, TD, 0);  // text
    gemm(x + TD, 2 * TD, P(11), TD, P(12), nullptr, 0, tmp1, HH, N, HH, TD, 0);  // audio
    encoder_finalize<<<N, 256, 0, stream>>>(hacc, tmp1, P(9), P(10), P(13), P(14),
                                            ntype, pos, P(16), P(17), P(18), P(19),
                                            P(15), h, N);

    // ---- RGAT layers
    for (int L = 0; L < 3; ++L) {
        int pb = 20 + L * 14;
        const float *Wq = P(pb), *Wk = P(pb + 1), *Wv = P(pb + 2);
        const float *rm1w = P(pb + 3), *rm1b = P(pb + 4);
        const float *rm2w = P(pb + 5), *rm2b = P(pb + 6), *sel = P(pb + 7);
        const float *Wsw = P(pb + 8), *Wsb = P(pb + 9);
        const float *crw = P(pb + 10), *crb = P(pb + 11);
        const float *lng = P(pb + 12), *lnb = P(pb + 13);

        fill(agg, (size_t)N * RR * HH, 0.f);
        for (int r = 0; r < RR; ++r) {
            gemm(h, HH, Wq + (size_t)r * HH * HH, HH, nullptr, nullptr, 0, Qb, HH, N, HH, HH, 0);
            gemm(h, HH, Wk + (size_t)r * HH * HH, HH, nullptr, nullptr, 0, Kb, HH, N, HH, HH, 0);
            gemm(h, HH, Wv + (size_t)r * HH * HH, HH, nullptr, nullptr, 0, Vb, HH, N, HH, HH, 0);
            fill(mb, (size_t)N * NHH, NEGINF);
            fill(zb, (size_t)N * NHH, 0.f);
            edge_score<<<eBlocks, 256, 0, stream>>>(src, dst, etype, E, r, Qb, Kb, score, mb);
            edge_exp<<<eBlocks, 256, 0, stream>>>(dst, etype, E, r, score, mb, zb);
            edge_agg<<<eBlocks, 256, 0, stream>>>(src, dst, etype, E, r, score, zb, Vb, agg);
        }
        // routing gate = softmax(rm2(relu(rm1(h))))
        gemm(h, HH, rm1w, HH, rm1b, nullptr, 0, tmp1, HH, N, HH, HH, GEMM_RELU);
        gemm(tmp1, HH, rm2w, HH, rm2b, nullptr, 0, gate, KK, N, KK, HH, 0);
        gate_softmax<<<(N + 255) / 256, 256, 0, stream>>>(gate, N);
        sew_softmax<<<1, 32, 0, stream>>>(sel, sew);
        // per-symptom chunk: s_k -> relu(Ws_k)*gate_k -> cross-accumulate
        for (int k = 0; k < KK; ++k) {
            sk_combine<<<(unsigned)(((size_t)N * HH + 255) / 256), 256, 0, stream>>>(agg, sew, k, tmp1, N);
            gemm(tmp1, HH, Wsw + (size_t)k * HH * HH, HH, Wsb + k * HH,
                 gate + k, KK, tmp2, HH, N, HH, HH, GEMM_RELU);
            int fl = (k > 0 ? GEMM_ACCUM : 0) | (k == KK - 1 ? GEMM_RELU : 0);
            gemm(tmp2, HH, crw + (size_t)k * HH, KK * HH,
                 (k == 0 ? crb : nullptr), nullptr, 0, hacc, HH, N, HH, HH, fl);
        }
        resid_ln<<<N, 256, 0, stream>>>(h, hacc, lng, lnb);
    }

    // ---- attention pooling (text / audio) + fuse + heads
    fill(mT, B, NEGINF); fill(mA, B, NEGINF);
    fill(zT, B, 0.f);    fill(zA, B, 0.f);
    fill(gcat, (size_t)B * 512, 0.f);
    att_scores<<<(N + 255) / 256, 256, 0, stream>>>(h, P(62), P(63), P(64), P(65),
                                                    ntype, batch, stb, sab, mT, mA, N);
    att_z<<<(N + 255) / 256, 256, 0, stream>>>(stb, sab, mT, mA, ntype, batch, zT, zA, N);
    att_accum<<<(unsigned)(((size_t)N * HH + 255) / 256), 256, 0, stream>>>(
        h, stb, sab, zT, zA, ntype, batch, gcat, N);
    gemm(gcat, 512, P(66), 512, P(67), nullptr, 0, fbuf, HH, B, HH, 512, 0);
    ln_relu_rows<<<B, 256, 0, stream>>>(fbuf, P(68), P(69), gfb);
    heads_kernel<<<B, 128, 0, stream>>>(gfb, P(70), P(71), P(72), P(73),
                                        P(74), P(75), P(76), P(77), (float*)d_out);
}